// LlamaTritonModel_18760417149197
// MI455X (gfx1250) — compile-verified
//
#include <hip/hip_runtime.h>
#include <hip/hip_bf16.h>
#include <math.h>

typedef __attribute__((ext_vector_type(2))) float v2f;
typedef __attribute__((ext_vector_type(8))) float v8f;

static constexpr int kT  = 128;
static constexpr int kD  = 1024;
static constexpr int kL  = 8;
static constexpr int kH  = 16;
static constexpr int kDH = 64;
static constexpr int kF  = 2816;
static constexpr int kV  = 32000;
static constexpr float kEps = 1e-5f;

// ---------------------------------------------------------------------------
// Embedding gather: X[t, :] = emb[ids[t], :]   (float4 coalesced)
// ---------------------------------------------------------------------------
__global__ void embed_kernel(const int* __restrict__ ids,
                             const float* __restrict__ emb,
                             float* __restrict__ X) {
    int t = blockIdx.x;
    int id = ids[t];
    const float4* src = (const float4*)(emb + (size_t)id * kD);
    float4* dst = (float4*)(X + (size_t)t * kD);
    for (int i = threadIdx.x; i < kD / 4; i += 256) dst[i] = src[i];
}

// ---------------------------------------------------------------------------
// Row-wise RMSNorm over D=1024: Y[t,:] = X[t,:] * rsqrt(mean(x^2)+eps) * w
// ---------------------------------------------------------------------------
__global__ void rmsnorm_kernel(const float* __restrict__ X,
                               const float* __restrict__ w,
                               float* __restrict__ Y) {
    int t = blockIdx.x;
    __shared__ float red[256];
    float ss = 0.f;
    for (int i = threadIdx.x; i < kD; i += 256) {
        float v = X[(size_t)t * kD + i];
        ss += v * v;
    }
    red[threadIdx.x] = ss;
    __syncthreads();
    for (int s = 128; s > 0; s >>= 1) {
        if (threadIdx.x < s) red[threadIdx.x] += red[threadIdx.x + s];
        __syncthreads();
    }
    float inv = rsqrtf(red[0] / (float)kD + kEps);
    for (int i = threadIdx.x; i < kD; i += 256)
        Y[(size_t)t * kD + i] = X[(size_t)t * kD + i] * inv * w[i];
}

// ---------------------------------------------------------------------------
// WMMA fp32 GEMM: C[128 x N] = A[128 x K] * B[K x N] (+ optional residual R)
// One wave computes a 16x32 output tile: two V_WMMA_F32_16X16X4_F32
// accumulators sharing a single A fragment per K-step of 4.
// B is indexed with 32-bit lane offsets from the uniform base so the backend
// can use SGPR-base + VGPR-offset (GVS) addressing in the hot loop.
//
// Fragment layouts per CDNA5 ISA 7.12.2 (32-bit, wave32):
//  A 16x4 : lanes 0-15 -> M=lane, {K=k0,k0+1}; lanes 16-31 -> M=lane-16, {k0+2,k0+3}
//  B 4x16 : lanes 0-15 -> N=lane, rows {k0,k0+1}; lanes 16-31 -> rows {k0+2,k0+3}
//  C 16x16: vgpr j -> M = j + 8*(lane>=16), N = lane&15
// ---------------------------------------------------------------------------
__global__ void gemm_wmma_kernel(const float* __restrict__ A,
                                 const float* __restrict__ B,
                                 const float* __restrict__ R,   // residual, may be null
                                 float* __restrict__ C,
                                 int K, int N) {
    int n0   = blockIdx.x * 32;
    int m0   = blockIdx.y * 16;
    int lane = threadIdx.x;       // 0..31, one wave per block
    int half = lane >> 4;         // 0: K pair {0,1}, 1: K pair {2,3}
    int l16  = lane & 15;

    const float* Arow = A + (size_t)(m0 + l16) * K + 2 * half;   // M row, K-pair base
    int bidx = (2 * half) * N + n0 + l16;                        // 32-bit lane offset into B

    v8f acc0 = {}, acc1 = {};
#pragma unroll 4
    for (int k = 0; k < K; k += 4) {
        // prefetch the weight stream ~64 K-rows ahead (global_prefetch_b8;
        // speculative prefetch past the end is dropped by hardware)
        __builtin_prefetch((const void*)(B + (bidx + (k + 64) * N)), 0, 1);

        v2f a = *(const v2f*)(Arow + k);        // 8-byte load, K pair
        int kb = bidx + k * N;
        v2f b0, b1;
        b0.x = B[kb];
        b0.y = B[kb + N];
        b1.x = B[kb + 16];
        b1.y = B[kb + N + 16];
        acc0 = __builtin_amdgcn_wmma_f32_16x16x4_f32(false, a, false, b0,
                                                     (short)0, acc0, false, false);
        acc1 = __builtin_amdgcn_wmma_f32_16x16x4_f32(false, a, false, b1,
                                                     (short)0, acc1, false, false);
    }

#pragma unroll
    for (int j = 0; j < 8; ++j) {
        size_t row = (size_t)(m0 + 8 * half + j) * N;
        size_t i0 = row + n0 + l16;
        size_t i1 = i0 + 16;
        float v0 = acc0[j];
        float v1 = acc1[j];
        if (R) { v0 += R[i0]; v1 += R[i1]; }
        C[i0] = v0;
        C[i1] = v1;
    }
}

// ---------------------------------------------------------------------------
// RoPE applied in place to Q and K. Layout [T, H*DH]; per head, pair (i, i+32).
// ---------------------------------------------------------------------------
__global__ void rope_kernel(float* __restrict__ Q, float* __restrict__ Kc) {
    int t = blockIdx.x;
    int j = threadIdx.x;          // 512 = H * DH/2
    int h = j >> 5;
    int i = j & 31;
    float inv = powf(10000.f, -(float)(2 * i) / (float)kDH);
    float ang = (float)t * inv;
    float c = cosf(ang), s = sinf(ang);
    size_t base = (size_t)t * kD + h * kDH;
    float q0 = Q[base + i], q1 = Q[base + i + 32];
    Q[base + i]      = q0 * c - q1 * s;
    Q[base + i + 32] = q1 * c + q0 * s;
    float k0 = Kc[base + i], k1 = Kc[base + i + 32];
    Kc[base + i]      = k0 * c - k1 * s;
    Kc[base + i + 32] = k1 * c + k0 * s;
}

// ---------------------------------------------------------------------------
// Causal attention, one block (128 threads) per (query token, head).
// scores -> masked softmax -> weighted sum of V.
// ---------------------------------------------------------------------------
__global__ void attn_kernel(const float* __restrict__ Q,
                            const float* __restrict__ Kc,
                            const float* __restrict__ Vc,
                            float* __restrict__ O) {
    int tq = blockIdx.x;
    int h  = blockIdx.y;
    int tid = threadIdx.x;        // 0..127

    __shared__ float qs[kDH];
    __shared__ float sc[kT];
    __shared__ float tmp[kT];

    if (tid < kDH) qs[tid] = Q[(size_t)tq * kD + h * kDH + tid];
    __syncthreads();

    float s = -1e9f;
    if (tid <= tq) {
        const float* kp = Kc + (size_t)tid * kD + h * kDH;
        float acc = 0.f;
#pragma unroll 8
        for (int d = 0; d < kDH; ++d) acc += qs[d] * kp[d];
        s = acc * 0.125f;         // 1/sqrt(64)
    }
    sc[tid]  = s;
    tmp[tid] = s;
    __syncthreads();
    for (int st = 64; st > 0; st >>= 1) {
        if (tid < st) tmp[tid] = fmaxf(tmp[tid], tmp[tid + st]);
        __syncthreads();
    }
    float mx = tmp[0];
    __syncthreads();
    float e = expf(sc[tid] - mx);
    sc[tid]  = e;
    tmp[tid] = e;
    __syncthreads();
    for (int st = 64; st > 0; st >>= 1) {
        if (tid < st) tmp[tid] += tmp[tid + st];
        __syncthreads();
    }
    float denom = tmp[0];
    __syncthreads();

    if (tid < kDH) {
        float acc = 0.f;
        for (int t2 = 0; t2 <= tq; ++t2)
            acc += sc[t2] * Vc[(size_t)t2 * kD + h * kDH + tid];
        O[(size_t)tq * kD + h * kDH + tid] = acc / denom;
    }
}

// ---------------------------------------------------------------------------
// SwiGLU elementwise: out = silu(g) * u
// ---------------------------------------------------------------------------
__global__ void silu_mul_kernel(const float* __restrict__ G,
                                const float* __restrict__ U,
                                float* __restrict__ Out, int n) {
    int i = blockIdx.x * 256 + threadIdx.x;
    if (i < n) {
        float g = G[i];
        Out[i] = (g / (1.f + expf(-g))) * U[i];
    }
}

// ---------------------------------------------------------------------------
// Final RMSNorm on the last token row -> xn[1024]
// ---------------------------------------------------------------------------
__global__ void final_norm_kernel(const float* __restrict__ X,
                                  const float* __restrict__ w,
                                  float* __restrict__ xn) {
    __shared__ float red[256];
    const float* x = X + (size_t)(kT - 1) * kD;
    float ss = 0.f;
    for (int i = threadIdx.x; i < kD; i += 256) {
        float v = x[i];
        ss += v * v;
    }
    red[threadIdx.x] = ss;
    __syncthreads();
    for (int s = 128; s > 0; s >>= 1) {
        if (threadIdx.x < s) red[threadIdx.x] += red[threadIdx.x + s];
        __syncthreads();
    }
    float inv = rsqrtf(red[0] / (float)kD + kEps);
    for (int i = threadIdx.x; i < kD; i += 256) xn[i] = x[i] * inv * w[i];
}

// ---------------------------------------------------------------------------
// Tied lm-head matvec: logits[v] = emb[v,:] . xn
// One wave32 per row; per-lane float4 -> coalesced global_load_b128.
// ---------------------------------------------------------------------------
__global__ void logits_kernel(const float* __restrict__ emb,
                              const float* __restrict__ xn,
                              float* __restrict__ out) {
    int wid  = (blockIdx.x * blockDim.x + threadIdx.x) >> 5;
    int lane = threadIdx.x & 31;
    if (wid >= kV) return;
    const float4* e = (const float4*)(emb + (size_t)wid * kD);
    const float4* x = (const float4*)xn;
    float acc = 0.f;
#pragma unroll
    for (int j = lane; j < kD / 4; j += 32) {   // 8 iterations
        float4 ev = e[j];
        float4 xv = x[j];
        acc += ev.x * xv.x + ev.y * xv.y + ev.z * xv.z + ev.w * xv.w;
    }
    for (int off = 16; off > 0; off >>= 1) acc += __shfl_xor(acc, off, 32);
    if (lane == 0) out[wid] = acc;
}

// ---------------------------------------------------------------------------
extern "C" void kernel_launch(void* const* d_in, const int* in_sizes, int n_in,
                              void* d_out, int out_size, void* d_ws, size_t ws_size,
                              hipStream_t stream) {
    const int*   ids      = (const int*)d_in[0];
    const float* emb      = (const float*)d_in[1];
    const float* Wq       = (const float*)d_in[2];
    const float* Wk       = (const float*)d_in[3];
    const float* Wv       = (const float*)d_in[4];
    const float* Wo       = (const float*)d_in[5];
    const float* Wg       = (const float*)d_in[6];
    const float* Wu       = (const float*)d_in[7];
    const float* Wd       = (const float*)d_in[8];
    const float* attn_n   = (const float*)d_in[9];
    const float* ffn_n    = (const float*)d_in[10];
    const float* norm_out = (const float*)d_in[11];

    float* ws = (float*)d_ws;
    float* X   = ws;                       // [T, D] residual stream
    float* Xn  = X   + (size_t)kT * kD;    // [T, D]
    float* Qb  = Xn  + (size_t)kT * kD;    // [T, D]
    float* Kb  = Qb  + (size_t)kT * kD;    // [T, D]
    float* Vb  = Kb  + (size_t)kT * kD;    // [T, D]
    float* AO  = Vb  + (size_t)kT * kD;    // [T, D]
    float* Gb  = AO  + (size_t)kT * kD;    // [T, F]
    float* Ub  = Gb  + (size_t)kT * kF;    // [T, F]
    float* Hb  = Ub  + (size_t)kT * kF;    // [T, F]
    float* xnl = Hb  + (size_t)kT * kF;    // [D]

    embed_kernel<<<kT, 256, 0, stream>>>(ids, emb, X);

    dim3 gD(kD / 32, kT / 16);   // 32 x 8 tiles of 16x32
    dim3 gF(kF / 32, kT / 16);   // 88 x 8 tiles of 16x32

    for (int l = 0; l < kL; ++l) {
        const float* wq = Wq + (size_t)l * kD * kD;
        const float* wk = Wk + (size_t)l * kD * kD;
        const float* wv = Wv + (size_t)l * kD * kD;
        const float* wo = Wo + (size_t)l * kD * kD;
        const float* wg = Wg + (size_t)l * kD * kF;
        const float* wu = Wu + (size_t)l * kD * kF;
        const float* wd = Wd + (size_t)l * kF * kD;

        // Attention block
        rmsnorm_kernel<<<kT, 256, 0, stream>>>(X, attn_n + (size_t)l * kD, Xn);
        gemm_wmma_kernel<<<gD, 32, 0, stream>>>(Xn, wq, nullptr, Qb, kD, kD);
        gemm_wmma_kernel<<<gD, 32, 0, stream>>>(Xn, wk, nullptr, Kb, kD, kD);
        gemm_wmma_kernel<<<gD, 32, 0, stream>>>(Xn, wv, nullptr, Vb, kD, kD);
        rope_kernel<<<kT, 512, 0, stream>>>(Qb, Kb);
        attn_kernel<<<dim3(kT, kH), 128, 0, stream>>>(Qb, Kb, Vb, AO);
        gemm_wmma_kernel<<<gD, 32, 0, stream>>>(AO, wo, X, X, kD, kD);  // fused residual

        // FFN block
        rmsnorm_kernel<<<kT, 256, 0, stream>>>(X, ffn_n + (size_t)l * kD, Xn);
        gemm_wmma_kernel<<<gF, 32, 0, stream>>>(Xn, wg, nullptr, Gb, kD, kF);
        gemm_wmma_kernel<<<gF, 32, 0, stream>>>(Xn, wu, nullptr, Ub, kD, kF);
        silu_mul_kernel<<<(kT * kF + 255) / 256, 256, 0, stream>>>(Gb, Ub, Hb, kT * kF);
        gemm_wmma_kernel<<<gD, 32, 0, stream>>>(Hb, wd, X, X, kF, kD);  // fused residual
    }

    final_norm_kernel<<<1, 256, 0, stream>>>(X, norm_out, xnl);
    logits_kernel<<<kV / 8, 256, 0, stream>>>(emb, xnl, (float*)d_out);
}